// SAGE_54640573940263
// MI455X (gfx1250) — compile-verified
//
#include <hip/hip_runtime.h>
#include <math.h>

typedef __attribute__((ext_vector_type(2))) float v2f;
typedef __attribute__((ext_vector_type(8))) float v8f;

#define NNODES 50000
#define NEDGES 800000
#define DIN0   384
#define HDIM   256
#define L2EPS  1e-12f

#define BM   128   // output rows per block
#define KBLK 16    // K tile
#define LDA  20    // A LDS row stride (floats): conflict-free & float4-aligned

// ---------------- elementwise / graph kernels ----------------

__global__ void k_log1p(const float* __restrict__ x, float* __restrict__ h, int n) {
  int i = blockIdx.x * blockDim.x + threadIdx.x;
  if (i < n) h[i] = log1pf(x[i]);
}

__global__ void k_degree(const int* __restrict__ dst, float* __restrict__ deg, int e) {
  int i = blockIdx.x * blockDim.x + threadIdx.x;
  if (i < e) atomicAdd(&deg[dst[i]], 1.0f);
}

// one thread per (edge, 4-feature chunk): coalesced float4 gather + 4 f32 atomics
__global__ void k_scatter(const float* __restrict__ h, const int* __restrict__ src,
                          const int* __restrict__ dst, float* __restrict__ agg,
                          int din, int e) {
  const int chunks = din >> 2;
  long long t = (long long)blockIdx.x * blockDim.x + threadIdx.x;
  if (t >= (long long)e * chunks) return;
  int ed = (int)(t / chunks);
  int c  = (int)(t - (long long)ed * chunks);
  int s = src[ed], d = dst[ed];
  float4 v = *((const float4*)(h + (long long)s * din) + c);
  float* ap = agg + (long long)d * din + (c << 2);
  atomicAdd(ap + 0, v.x);
  atomicAdd(ap + 1, v.y);
  atomicAdd(ap + 2, v.z);
  atomicAdd(ap + 3, v.w);
}

__global__ void k_normagg(float* __restrict__ agg, const float* __restrict__ deg, int din) {
  const int chunks = din >> 2;
  long long t = (long long)blockIdx.x * blockDim.x + threadIdx.x;
  if (t >= (long long)NNODES * chunks) return;
  int i = (int)(t / chunks);
  int c = (int)(t - (long long)i * chunks);
  float inv = 1.0f / fmaxf(deg[i], 1.0f);
  float4* p = (float4*)(agg + (long long)i * din) + c;
  float4 v = *p;
  v.x *= inv; v.y *= inv; v.z *= inv; v.w *= inv;
  *p = v;
}

// ---------------- fused SAGE layer GEMM ----------------
// out[m, 0:256] = A[m,:] @ Ws + An[m,:] @ Wn + bias, then (optional) relu + row L2-norm.
// Block = 8 wave32 waves, tile 128 x 256 (full H -> L2-norm is block-local).
// Wave w owns rows [w*16, w*16+16), 16 n-subtiles, K stepped by v_wmma_f32_16x16x4_f32.

__global__ __launch_bounds__(256) void k_sage_gemm(
    const float* __restrict__ A,    // [n, din] self features
    const float* __restrict__ An,   // [n, din] mean-aggregated neighbor features
    const float* __restrict__ Ws,   // [din, 256]
    const float* __restrict__ Wn,   // [din, 256]
    const float* __restrict__ bias, // [256]
    float* __restrict__ out,        // [n, 256]
    int n, int din, int fuse)
{
  __shared__ float As [BM * LDA];
  __shared__ float Ans[BM * LDA];
  __shared__ float Bs [(KBLK / 4) * HDIM * 4];  // swizzled: [(g*256+n)*4 + (k&3)]
  __shared__ float Bn [(KBLK / 4) * HDIM * 4];

  const int tid   = threadIdx.x;
  const int lane  = tid & 31;
  const int wid   = tid >> 5;
  const int l15   = lane & 15;
  const int khalf = (lane >> 4) << 1;   // 0 for lanes 0-15 (K,K+1), 2 for lanes 16-31 (K+2,K+3)
  const int m0    = wid * 16;
  const int rowblk = blockIdx.x * BM;

  v8f acc[16];
#pragma unroll
  for (int i = 0; i < 16; ++i)
    acc[i] = (v8f){0.f, 0.f, 0.f, 0.f, 0.f, 0.f, 0.f, 0.f};

  for (int kb = 0; kb < din; kb += KBLK) {
    // ---- stage A / An: 128 rows x 16 k, coalesced float4 loads ----
#pragma unroll
    for (int p = 0; p < 2; ++p) {
      int rl = p * 64 + (tid >> 2);
      int kc = (tid & 3) * 4;
      int gr = rowblk + rl;
      float4 va = make_float4(0.f, 0.f, 0.f, 0.f);
      float4 vn = va;
      if (gr < n) {
        va = *(const float4*)(A  + (long long)gr * din + kb + kc);
        vn = *(const float4*)(An + (long long)gr * din + kb + kc);
      }
      *(float4*)&As [rl * LDA + kc] = va;
      *(float4*)&Ans[rl * LDA + kc] = vn;
    }
    // ---- stage Ws / Wn slice with WMMA-frag swizzle ----
#pragma unroll
    for (int p = 0; p < 4; ++p) {
      int kk = p * 4 + (tid >> 6);     // 0..15 within K tile
      int nn = (tid & 63) * 4;
      float4 vs = *(const float4*)(Ws + (long long)(kb + kk) * HDIM + nn);
      float4 vn = *(const float4*)(Wn + (long long)(kb + kk) * HDIM + nn);
      int g = kk >> 2, kr = kk & 3;
      Bs[((g * HDIM + nn + 0) << 2) + kr] = vs.x;
      Bs[((g * HDIM + nn + 1) << 2) + kr] = vs.y;
      Bs[((g * HDIM + nn + 2) << 2) + kr] = vs.z;
      Bs[((g * HDIM + nn + 3) << 2) + kr] = vs.w;
      Bn[((g * HDIM + nn + 0) << 2) + kr] = vn.x;
      Bn[((g * HDIM + nn + 1) << 2) + kr] = vn.y;
      Bn[((g * HDIM + nn + 2) << 2) + kr] = vn.z;
      Bn[((g * HDIM + nn + 3) << 2) + kr] = vn.w;
    }
    __syncthreads();

#pragma unroll
    for (int g = 0; g < KBLK / 4; ++g) {
      int k = g * 4;
      v2f as = *(const v2f*)&As [(m0 + l15) * LDA + k + khalf];
      v2f an = *(const v2f*)&Ans[(m0 + l15) * LDA + k + khalf];
#pragma unroll
      for (int ns = 0; ns < 16; ++ns) {
        v2f bs = *(const v2f*)&Bs[((g * HDIM + ns * 16 + l15) << 2) + khalf];
        v2f bn = *(const v2f*)&Bn[((g * HDIM + ns * 16 + l15) << 2) + khalf];
        acc[ns] = __builtin_amdgcn_wmma_f32_16x16x4_f32(
            false, as, false, bs, (short)0, acc[ns], false, false);
        acc[ns] = __builtin_amdgcn_wmma_f32_16x16x4_f32(
            false, an, false, bn, (short)0, acc[ns], false, false);
      }
    }
    __syncthreads();
  }

  // ---- epilogue ----
  const int rbase = rowblk + m0 + ((lane >> 4) << 3);  // + r gives this lane's row for VGPR r

  if (fuse) {
    float ss[8];
#pragma unroll
    for (int r = 0; r < 8; ++r) ss[r] = 0.f;
#pragma unroll
    for (int ns = 0; ns < 16; ++ns) {
      float bv = bias[ns * 16 + l15];
#pragma unroll
      for (int r = 0; r < 8; ++r) {
        float v = acc[ns][r] + bv;
        v = fmaxf(v, 0.f);
        acc[ns][r] = v;
        ss[r] += v * v;
      }
    }
    // reduce sum-of-squares across the 16 lanes sharing each row (half-wave)
#pragma unroll
    for (int r = 0; r < 8; ++r) {
      float s = ss[r];
      s += __shfl_xor(s, 1);
      s += __shfl_xor(s, 2);
      s += __shfl_xor(s, 4);
      s += __shfl_xor(s, 8);
      ss[r] = 1.0f / fmaxf(sqrtf(s), L2EPS);
    }
#pragma unroll
    for (int ns = 0; ns < 16; ++ns) {
#pragma unroll
      for (int r = 0; r < 8; ++r) {
        int row = rbase + r;
        if (row < n)
          out[(long long)row * HDIM + ns * 16 + l15] = acc[ns][r] * ss[r];
      }
    }
  } else {
#pragma unroll
    for (int ns = 0; ns < 16; ++ns) {
      float bv = bias[ns * 16 + l15];
#pragma unroll
      for (int r = 0; r < 8; ++r) {
        int row = rbase + r;
        if (row < n)
          out[(long long)row * HDIM + ns * 16 + l15] = acc[ns][r] + bv;
      }
    }
  }
}

// ---------------- host orchestration ----------------

extern "C" void kernel_launch(void* const* d_in, const int* in_sizes, int n_in,
                              void* d_out, int out_size, void* d_ws, size_t ws_size,
                              hipStream_t stream) {
  const float* x   = (const float*)d_in[0];
  const int*   src = (const int*)d_in[1];
  const int*   dst = (const int*)d_in[2];
  const float* Ws0 = (const float*)d_in[3];
  const float* Wn0 = (const float*)d_in[4];
  const float* b0  = (const float*)d_in[5];
  const float* Ws1 = (const float*)d_in[6];
  const float* Wn1 = (const float*)d_in[7];
  const float* b1  = (const float*)d_in[8];
  const float* Ws2 = (const float*)d_in[9];
  const float* Wn2 = (const float*)d_in[10];
  const float* b2  = (const float*)d_in[11];
  float* out = (float*)d_out;

  float* ws  = (float*)d_ws;
  float* h0  = ws;                                  // N x 384 (also reused for h2)
  float* agg = h0  + (size_t)NNODES * DIN0;         // N x 384 (max din)
  float* h1  = agg + (size_t)NNODES * DIN0;         // N x 256
  float* deg = h1  + (size_t)NNODES * HDIM;         // N

  const int gemm_blocks = (NNODES + BM - 1) / BM;

  // h0 = log1p(x)
  {
    int total = NNODES * DIN0;
    k_log1p<<<(total + 255) / 256, 256, 0, stream>>>(x, h0, total);
  }
  // degree (shared by all layers)
  hipMemsetAsync(deg, 0, NNODES * sizeof(float), stream);
  k_degree<<<(NEDGES + 255) / 256, 256, 0, stream>>>(dst, deg, NEDGES);

  // ---- layer 0: din=384, relu + l2norm -> h1 ----
  hipMemsetAsync(agg, 0, (size_t)NNODES * DIN0 * sizeof(float), stream);
  {
    long long total = (long long)NEDGES * (DIN0 / 4);
    k_scatter<<<(int)((total + 255) / 256), 256, 0, stream>>>(h0, src, dst, agg, DIN0, NEDGES);
  }
  {
    long long total = (long long)NNODES * (DIN0 / 4);
    k_normagg<<<(int)((total + 255) / 256), 256, 0, stream>>>(agg, deg, DIN0);
  }
  k_sage_gemm<<<gemm_blocks, 256, 0, stream>>>(h0, agg, Ws0, Wn0, b0, h1, NNODES, DIN0, 1);

  // ---- layer 1: din=256, relu + l2norm -> h2 (stored in h0 buffer) ----
  hipMemsetAsync(agg, 0, (size_t)NNODES * HDIM * sizeof(float), stream);
  {
    long long total = (long long)NEDGES * (HDIM / 4);
    k_scatter<<<(int)((total + 255) / 256), 256, 0, stream>>>(h1, src, dst, agg, HDIM, NEDGES);
  }
  {
    long long total = (long long)NNODES * (HDIM / 4);
    k_normagg<<<(int)((total + 255) / 256), 256, 0, stream>>>(agg, deg, HDIM);
  }
  k_sage_gemm<<<gemm_blocks, 256, 0, stream>>>(h1, agg, Ws1, Wn1, b1, h0, NNODES, HDIM, 1);

  // ---- layer 2: din=256, plain -> d_out ----
  hipMemsetAsync(agg, 0, (size_t)NNODES * HDIM * sizeof(float), stream);
  {
    long long total = (long long)NEDGES * (HDIM / 4);
    k_scatter<<<(int)((total + 255) / 256), 256, 0, stream>>>(h0, src, dst, agg, HDIM, NEDGES);
  }
  {
    long long total = (long long)NNODES * (HDIM / 4);
    k_normagg<<<(int)((total + 255) / 256), 256, 0, stream>>>(agg, deg, HDIM);
  }
  k_sage_gemm<<<gemm_blocks, 256, 0, stream>>>(h0, agg, Ws2, Wn2, b2, out, NNODES, HDIM, 0);
}